// Loss_57415122813634
// MI455X (gfx1250) — compile-verified
//
#include <hip/hip_runtime.h>

// Problem geometry (fixed by the reference: B=2, C=3, H=W=256, scales 1,4,...,199)
#define HH 256
#define WW 256
#define CHN 3
#define NSCALE 67                 // range(1, 200, 3)
#define NDIR 3
#define UNITS_PER_BATCH (NSCALE * NDIR)          // 201
#define NBATCH 2
#define SPLIT 4                                   // blocks per (b,scale,dir) unit
#define BLOCKS_PER_BATCH (UNITS_PER_BATCH * SPLIT) // 804
#define NBLK (NBATCH * BLOCKS_PER_BATCH)           // 1608
#define PLANE (HH * WW)

// Workspace float offsets
#define OFF_S1  0
#define OFF_S2  (NBLK)
#define OFF_PC  (2 * NBLK)
#define OFF_SAP (3 * NBLK)
#define OFF_T1  (4 * NBLK)
#define OFF_SC  (5 * NBLK)   // +0..1: NormConst c, +2..3: roiSum, +4..5: posCnt, +6..7: sum|difPrd|

__device__ __forceinline__ float blockReduceF(float v, float* sm) {
    const int t = threadIdx.x;
    __syncthreads();               // protect sm reuse across back-to-back calls
    sm[t] = v;
    __syncthreads();
    for (int o = 128; o >= 1; o >>= 1) {
        if (t < o) sm[t] += sm[t + o];
        __syncthreads();
    }
    return sm[0];
}

__device__ __forceinline__ double blockReduceD(double v, double* sm) {
    const int t = threadIdx.x;
    __syncthreads();
    sm[t] = v;
    __syncthreads();
    for (int o = 128; o >= 1; o >>= 1) {
        if (t < o) sm[t] += sm[t + o];
        __syncthreads();
    }
    return sm[0];
}

// Decode block -> (batch, split-part, dy, dx, valid-region geometry)
__device__ __forceinline__ void decodeBlock(int blk, int& b, int& part, int& dy, int& dx,
                                            int& wv, int& hv, int& shift) {
    b = blk / BLOCKS_PER_BATCH;
    const int rem  = blk % BLOCKS_PER_BATCH;
    const int unit = rem / SPLIT;
    part           = rem % SPLIT;
    const int sIdx = unit / NDIR;
    const int dir  = unit % NDIR;
    const int s    = 1 + 3 * sIdx;
    // reference dir order: (0,+s), (+s,0), (+s,+s)
    dy = (dir == 0) ? 0 : s;
    dx = (dir == 1) ? 0 : s;
    wv = WW - dx;                    // only in-bounds pixels contribute (OOB terms are exactly 0)
    hv = HH - dy;
    shift = dy * WW + dx;
}

// Pass A: per-block partial sums of S1, S2, posCnt, sum|difPrd|
__global__ __launch_bounds__(256) void passA_kernel(const float* __restrict__ Pred,
                                                    const float* __restrict__ GT,
                                                    const float* __restrict__ ROI,
                                                    float* __restrict__ ws) {
    __shared__ float sm[256];
    int b, part, dy, dx, wv, hv, shift;
    decodeBlock(blockIdx.x, b, part, dy, dx, wv, hv, shift);

    const float* __restrict__ roiB = ROI  + b * PLANE;
    const float* __restrict__ P0   = Pred + (b * CHN + 0) * PLANE;
    const float* __restrict__ P1   = Pred + (b * CHN + 1) * PLANE;
    const float* __restrict__ P2   = Pred + (b * CHN + 2) * PLANE;
    const float* __restrict__ G0   = GT   + (b * CHN + 0) * PLANE;
    const float* __restrict__ G1   = GT   + (b * CHN + 1) * PLANE;
    const float* __restrict__ G2   = GT   + (b * CHN + 2) * PLANE;

    float s1 = 0.f, s2 = 0.f, sap = 0.f, pc = 0.f;
    for (int y = part; y < hv; y += SPLIT) {            // rows strided across the split blocks
        const int row = y * WW;
        for (int x = (int)threadIdx.x; x < wv; x += 256) {  // coalesced across threads
            const int idx = row + x;
            const float roi = roiB[idx] * roiB[idx + shift];
            if (roi != 0.0f) {
                const float dPv[CHN] = { roi * (P0[idx] - P0[idx + shift]),
                                         roi * (P1[idx] - P1[idx + shift]),
                                         roi * (P2[idx] - P2[idx + shift]) };
                const float dGv[CHN] = { roi * (G0[idx] - G0[idx + shift]),
                                         roi * (G1[idx] - G1[idx + shift]),
                                         roi * (G2[idx] - G2[idx + shift]) };
#pragma unroll
                for (int ch = 0; ch < CHN; ++ch) {
                    const float dP = dPv[ch];
                    const float dG = dGv[ch];
                    const float ratio = dP / (dG + 1e-5f);   // Rat = relu(ratio)
                    const float aG = fabsf(dG);
                    if (ratio > 0.0f) { s1 += aG * ratio; s2 += aG; }
                    if (dG > 0.0f) pc += 1.0f;
                    sap += fabsf(dP);
                }
            }
        }
    }
    const float r1 = blockReduceF(s1,  sm);
    const float r2 = blockReduceF(s2,  sm);
    const float r3 = blockReduceF(pc,  sm);
    const float r4 = blockReduceF(sap, sm);
    if (threadIdx.x == 0) {
        ws[OFF_S1  + blockIdx.x] = r1;
        ws[OFF_S2  + blockIdx.x] = r2;
        ws[OFF_PC  + blockIdx.x] = r3;
        ws[OFF_SAP + blockIdx.x] = r4;
    }
}

// Reduce pass-A partials (deterministic, double accumulation), compute NormConst + scalars
__global__ __launch_bounds__(256) void reduceA_kernel(const float* __restrict__ ROI,
                                                      float* __restrict__ ws,
                                                      float* __restrict__ out) {
    __shared__ double dsm[256];
    const int t = threadIdx.x;
    for (int b = 0; b < NBATCH; ++b) {
        const int base = b * BLOCKS_PER_BATCH;
        double a1 = 0, a2 = 0, ap = 0, as = 0;
        for (int i = t; i < BLOCKS_PER_BATCH; i += 256) {
            a1 += (double)ws[OFF_S1  + base + i];
            a2 += (double)ws[OFF_S2  + base + i];
            ap += (double)ws[OFF_PC  + base + i];
            as += (double)ws[OFF_SAP + base + i];
        }
        a1 = blockReduceD(a1, dsm);
        a2 = blockReduceD(a2, dsm);
        ap = blockReduceD(ap, dsm);
        as = blockReduceD(as, dsm);

        double rsum = 0;
        const float* __restrict__ roiB = ROI + b * PLANE;
        for (int i = t; i < PLANE; i += 256) rsum += (double)roiB[i];
        rsum = blockReduceD(rsum, dsm);

        if (t == 0) {
            const float c = (float)(a1 / (a2 + 1e-4));   // NormConst (stop-gradient scalar)
            ws[OFF_SC + 0 + b] = c;
            ws[OFF_SC + 2 + b] = (float)rsum;
            ws[OFF_SC + 4 + b] = (float)ap;
            ws[OFF_SC + 6 + b] = (float)as;
            out[1 + b] = c;                              // NormConst output
        }
    }
}

// Pass B: per-block partial sums of |difGT - difPrd / safe_c|
__global__ __launch_bounds__(256) void passB_kernel(const float* __restrict__ Pred,
                                                    const float* __restrict__ GT,
                                                    const float* __restrict__ ROI,
                                                    float* __restrict__ ws) {
    __shared__ float sm[256];
    int b, part, dy, dx, wv, hv, shift;
    decodeBlock(blockIdx.x, b, part, dy, dx, wv, hv, shift);

    const float c   = ws[OFF_SC + b];
    const float inv = (c > 1e-4f) ? (1.0f / c) : 1.0f;   // 1 / safe_c

    const float* __restrict__ roiB = ROI  + b * PLANE;
    const float* __restrict__ P0   = Pred + (b * CHN + 0) * PLANE;
    const float* __restrict__ P1   = Pred + (b * CHN + 1) * PLANE;
    const float* __restrict__ P2   = Pred + (b * CHN + 2) * PLANE;
    const float* __restrict__ G0   = GT   + (b * CHN + 0) * PLANE;
    const float* __restrict__ G1   = GT   + (b * CHN + 1) * PLANE;
    const float* __restrict__ G2   = GT   + (b * CHN + 2) * PLANE;

    float acc = 0.f;
    for (int y = part; y < hv; y += SPLIT) {
        const int row = y * WW;
        for (int x = (int)threadIdx.x; x < wv; x += 256) {
            const int idx = row + x;
            const float roi = roiB[idx] * roiB[idx + shift];
            if (roi != 0.0f) {
                const float dP0 = roi * (P0[idx] - P0[idx + shift]);
                const float dP1 = roi * (P1[idx] - P1[idx + shift]);
                const float dP2 = roi * (P2[idx] - P2[idx + shift]);
                const float dG0 = roi * (G0[idx] - G0[idx + shift]);
                const float dG1 = roi * (G1[idx] - G1[idx + shift]);
                const float dG2 = roi * (G2[idx] - G2[idx + shift]);
                acc += fabsf(dG0 - dP0 * inv);
                acc += fabsf(dG1 - dP1 * inv);
                acc += fabsf(dG2 - dP2 * inv);
            }
        }
    }
    const float r = blockReduceF(acc, sm);
    if (threadIdx.x == 0) ws[OFF_T1 + blockIdx.x] = r;
}

// Final: reduce T1 partials, apply term1/2/3 gating, write loss
__global__ __launch_bounds__(256) void finalize_kernel(const float* __restrict__ ws,
                                                       float* __restrict__ out) {
    __shared__ double dsm[256];
    __shared__ double t1s[NBATCH];
    const int t = threadIdx.x;
    for (int b = 0; b < NBATCH; ++b) {
        const int base = b * BLOCKS_PER_BATCH;
        double a = 0;
        for (int i = t; i < BLOCKS_PER_BATCH; i += 256) a += (double)ws[OFF_T1 + base + i];
        a = blockReduceD(a, dsm);
        if (t == 0) t1s[b] = a;
    }
    __syncthreads();
    if (t == 0) {
        const double N = (double)(NSCALE * NDIR * CHN) * (double)PLANE;  // 603 * 65536
        double loss = 0.0;
        for (int b = 0; b < NBATCH; ++b) {
            const float c       = ws[OFF_SC + 0 + b];
            const float roiSum  = ws[OFF_SC + 2 + b];
            const float pcnt    = ws[OFF_SC + 4 + b];
            const float sap     = ws[OFF_SC + 6 + b];
            const double term1  = (c > 1e-4f) ? (t1s[b] / N) : 0.0;
            const float mean_prd = sap / pcnt;
            const bool big = roiSum > 200.0f;
            const double term2 = (big && (mean_prd > 30.0f) && (c > 10.0f))
                                     ? (double)(mean_prd - 30.0f) : 0.0;
            const float fact = 0.1f / (c + 0.001f);
            const double term3 = (big && (mean_prd < 2.0f) && (c < 0.1f))
                                     ? (double)((0.2f - mean_prd) * fact) : 0.0;
            loss += term1 + term2 + term3;
        }
        out[0] = (float)loss;
    }
}

extern "C" void kernel_launch(void* const* d_in, const int* in_sizes, int n_in,
                              void* d_out, int out_size, void* d_ws, size_t ws_size,
                              hipStream_t stream) {
    (void)in_sizes; (void)n_in; (void)out_size; (void)ws_size;
    const float* Pred = (const float*)d_in[0];   // [2,3,256,256] f32
    const float* GT   = (const float*)d_in[1];   // [2,3,256,256] f32
    const float* ROI  = (const float*)d_in[2];   // [2,1,256,256] f32
    float* out = (float*)d_out;                  // [loss, NormConst0, NormConst1]
    float* ws  = (float*)d_ws;                   // needs (5*NBLK + 8) floats ≈ 32 KB

    passA_kernel<<<NBLK, 256, 0, stream>>>(Pred, GT, ROI, ws);
    reduceA_kernel<<<1, 256, 0, stream>>>(ROI, ws, out);
    passB_kernel<<<NBLK, 256, 0, stream>>>(Pred, GT, ROI, ws);
    finalize_kernel<<<1, 256, 0, stream>>>(ws, out);
}